// Angles2Backbone_41944650612748
// MI455X (gfx1250) — compile-verified
//
#include <hip/hip_runtime.h>
#include <cstdint>

// ---------------------------------------------------------------------------
// Angles2Backbone on MI455X (gfx1250, wave32).
//
// Structure: 1 block per batch chain (B=256 blocks, 256 threads = 8 waves).
//  1) TDM tensor_load_to_lds stages the 2*L angle row (16KB) into LDS.
//  2) Each thread builds+folds its 24-matrix segment (affine 3x4, VALU).
//  3) 8-step Kogge-Stone scan over 256 segment totals. The combine products
//     are done on the matrix pipe: V_WMMA_F32_16X16X4_F32 computes four 4x4
//     products per instruction (A = 4 stacked left matrices (16x4),
//     B = 4 side-by-side right matrices (4x16), diagonal D blocks = results).
//     Ping-pong LDS buffers, identity matrix at slot 256 for i < stride.
//  4) Threads re-walk their segment applying the exclusive carry, write
//     masked positions into an LDS coord tile.
//  5) TDM tensor_store_from_lds DMAs the 72KB coord tile to HBM.
// ---------------------------------------------------------------------------

typedef unsigned int v4u __attribute__((ext_vector_type(4)));
typedef int          v4i __attribute__((ext_vector_type(4)));
typedef int          v8i __attribute__((ext_vector_type(8)));
typedef float        v2f __attribute__((ext_vector_type(2)));
typedef float        v8f __attribute__((ext_vector_type(8)));

struct TdmDesc { v4u g0; v8i g1; v4i g2; v4i g3; };

// Build a D# for a 1-D contiguous tile of n 4-byte elements (ISA 8.3-8.6).
__device__ __forceinline__ TdmDesc tdm_desc_1d(uint64_t gaddr, unsigned lds_off,
                                               unsigned n) {
  TdmDesc d;
  // group0: count=1 valid user desc, lds_addr, 57-bit global addr, type=2
  d.g0[0] = 1u;
  d.g0[1] = lds_off;
  d.g0[2] = (unsigned)(gaddr & 0xFFFFFFFFull);
  d.g0[3] = (unsigned)((gaddr >> 32) & 0x1FFFFFFull) | (2u << 30);
  const unsigned td0 = n, td1 = 1u, tile0 = n, tile1 = 1u, tile2 = 1u;
  const uint64_t s0 = n, s1 = n;
  // group1: data_size=4B (code 2), no multicast/pad/iterate
  d.g1[0] = (int)(2u << 16);
  d.g1[1] = (int)((td0 & 0xFFFFu) << 16);
  d.g1[2] = (int)(((td0 >> 16) & 0xFFFFu) | ((td1 & 0xFFFFu) << 16));
  d.g1[3] = (int)(((td1 >> 16) & 0xFFFFu) | ((tile0 & 0xFFFFu) << 16));
  d.g1[4] = (int)((tile1 & 0xFFFFu) | ((tile2 & 0xFFFFu) << 16));
  d.g1[5] = (int)(s0 & 0xFFFFFFFFull);
  d.g1[6] = (int)(((s0 >> 32) & 0xFFFFull) | ((unsigned)(s1 & 0xFFFFull) << 16));
  d.g1[7] = (int)((s1 >> 16) & 0xFFFFFFFFull);
  // group2: tensor_dim2=1, tensor_dim3=1, dim2 stride, tile_dim3=1
  d.g2[0] = 1; d.g2[1] = 1;
  d.g2[2] = (int)(s1 & 0xFFFFFFFFull);
  d.g2[3] = (int)(((s1 >> 32) & 0xFFFFull) | (1u << 16));
  // group3: dim3 stride, tensor_dim4=1, tile_dim4=1
  d.g3[0] = (int)(s1 & 0xFFFFFFFFull);
  d.g3[1] = (int)(((s1 >> 32) & 0xFFFFull) | (1u << 16));
  d.g3[2] = (int)(1u << 16);
  d.g3[3] = 0;
  return d;
}

// A (3x4 affine, row-major, implicit bottom row 0001):  A <- A * M
__device__ __forceinline__ void compose(float a[12], const float m[12]) {
#pragma unroll
  for (int i = 0; i < 3; ++i) {
    const float x = a[i * 4 + 0], y = a[i * 4 + 1];
    const float z = a[i * 4 + 2], w = a[i * 4 + 3];
    a[i * 4 + 0] = fmaf(x, m[0], fmaf(y, m[4], z * m[8]));
    a[i * 4 + 1] = fmaf(x, m[1], fmaf(y, m[5], z * m[9]));
    a[i * 4 + 2] = fmaf(x, m[2], fmaf(y, m[6], z * m[10]));
    a[i * 4 + 3] = fmaf(x, m[3], fmaf(y, m[7], fmaf(z, m[11], w)));
  }
}

__device__ __forceinline__ void stepk(float A[12], float th, float ca, float sa,
                                      float R) {
  const float ct = cosf(th), st = sinf(th);
  const float m[12] = {-ca,     -sa,     0.f, -R * ca,
                       sa * ct, -ca * ct, -st, R * (sa * ct),
                       sa * st, -ca * st,  ct, R * (sa * st)};
  compose(A, m);
}

__device__ __forceinline__ void emit_pos(const float A[12], int t, float* coords,
                                         int vlim) {
  const bool v = t < vlim;
  coords[3 * t + 0] = v ? A[3]  : 0.f;
  coords[3 * t + 1] = v ? A[7]  : 0.f;
  coords[3 * t + 2] = v ? A[11] : 0.f;
}

template <bool EMIT>
__device__ __forceinline__ void do_residue(float A[12], int l,
                                           const float* phiS, const float* psiS,
                                           float ca0, float sa0, float ca1,
                                           float sa1, float ca2, float sa2,
                                           float* coords, int vlim) {
  // k=0: theta = psi_prev ; t==0 is the identity matrix (reference M[0]=eye)
  if (l == 0) {
    if (EMIT) emit_pos(A, 0, coords, vlim);
  } else {
    stepk(A, psiS[l - 1], ca0, sa0, 1.33f);
    if (EMIT) emit_pos(A, 3 * l, coords, vlim);
  }
  // k=1: theta = omega = pi
  stepk(A, 3.14159265358979323846f, ca1, sa1, 1.46f);
  if (EMIT) emit_pos(A, 3 * l + 1, coords, vlim);
  // k=2: theta = phi
  stepk(A, phiS[l], ca2, sa2, 1.525f);
  if (EMIT) emit_pos(A, 3 * l + 2, coords, vlim);
}

__global__ __launch_bounds__(256) void angles2backbone_scan_kernel(
    const float* __restrict__ inp, const int* __restrict__ lens,
    float* __restrict__ out) {
  constexpr int L = 2048;      // residues
  constexpr int NRES = 8;      // residues per thread (3L / 256 / 3)
  constexpr int IDSLOT = 256;  // identity slot in scan buffers

  __shared__ __align__(16) float s_ang[2 * L];  // [phi | psi] : 16 KB
  __shared__ __align__(16) float s_uni[9 * L];  // scan ping-pong ∪ coords: 72KB

  float* const scan0  = s_uni;               // 257 * 16 floats
  float* const scan1  = s_uni + 257 * 16;    // 257 * 16 floats
  float* const coords = s_uni;               // overwrites scan after carries

  const int tid = threadIdx.x;
  const int b   = blockIdx.x;

  // -- 1) TDM: stage angles[b, :, :] (2L contiguous f32) into LDS ----------
  if (tid < 32) {  // wave 0 only: TDM issues once per wave, EXEC ignored
    const uint64_t ga = (uint64_t)(uintptr_t)(inp + (size_t)b * (2 * L));
    const unsigned lo = (unsigned)(uintptr_t)(void*)s_ang;  // flat low32 = LDS off
    TdmDesc d = tdm_desc_1d(ga, lo, 2u * L);
    __builtin_amdgcn_tensor_load_to_lds(d.g0, d.g1, d.g2, d.g3, (v8i)0, 0);
    __builtin_amdgcn_s_wait_tensorcnt(0);
  }
  if (tid == 0 || tid == 1) {  // identity matrices at slot 256 of both buffers
    float* p = (tid == 0 ? scan0 : scan1) + IDSLOT * 16;
#pragma unroll
    for (int e = 0; e < 16; ++e) p[e] = 0.f;
    p[0] = p[5] = p[10] = p[15] = 1.f;
  }
  __syncthreads();

  const float* phiS = s_ang;
  const float* psiS = s_ang + L;
  const float ca0 = cosf(2.1186f), sa0 = sinf(2.1186f);
  const float ca1 = cosf(1.9391f), sa1 = sinf(1.9391f);
  const float ca2 = cosf(1.9111f), sa2 = sinf(1.9111f);

  // -- 2) per-thread segment fold (24 matrices) -----------------------------
  float A[12] = {1.f, 0.f, 0.f, 0.f, 0.f, 1.f, 0.f, 0.f, 0.f, 0.f, 1.f, 0.f};
  for (int rj = 0; rj < NRES; ++rj)
    do_residue<false>(A, tid * NRES + rj, phiS, psiS, ca0, sa0, ca1, sa1, ca2,
                      sa2, nullptr, 0);
  {
    float* x = scan0 + tid * 16;
#pragma unroll
    for (int e = 0; e < 12; ++e) x[e] = A[e];
    x[12] = 0.f; x[13] = 0.f; x[14] = 0.f; x[15] = 1.f;
  }

  // -- 3) Kogge-Stone scan over 256 segment totals via V_WMMA_F32_16X16X4 --
  // Per wave, per step: 8 WMMAs, each producing four 4x4 products
  // Q_{i-s} x P_i on its diagonal blocks.
  const int lane   = tid & 31;
  const int wave   = tid >> 5;
  const bool hiHalf = lane >= 16;
  const int lq  = hiHalf ? lane - 16 : lane;  // 0..15
  const int mj  = lq >> 2;                    // which of 4 matrices
  const int rA  = lq & 3;                     // row in A / col in B
  const int K0  = hiHalf ? 2 : 0;             // K pair held by this lane half
  const bool doStore = (lane < 8) || (lane >= 24);  // lanes holding diag cols
  const int selj = (lane < 8) ? (lane >> 2) : (2 + ((lane - 24) >> 2));
  const int colD = lane & 3;
  const bool hiD = ((lane >> 2) & 1) != 0;  // block rows in d[4..7] vs d[0..3]

  for (int st = 0; st < 8; ++st) {
    const int s = 1 << st;
    const bool odd = (st & 1) != 0;
    const float* src = odd ? scan1 : scan0;   // ternary select, no ptr array
    float* dst       = odd ? scan0 : scan1;
    __syncthreads();  // previous step's writes visible
#pragma unroll
    for (int q = 0; q < 8; ++q) {
      const int rightBase = wave * 32 + q * 4;
      const int myRight = rightBase + mj;
      const int myLeft  = (myRight >= s) ? (myRight - s) : IDSLOT;
      // A(M=4*mj+rA, K): left matrix element [rA][K0], [rA][K0+1]
      const float* ap = src + myLeft * 16 + rA * 4 + K0;
      v2f av; av.x = ap[0]; av.y = ap[1];
      // B(K, N=4*mj+rA): right matrix element [K0][rA], [K0+1][rA]
      const float* bp = src + myRight * 16 + K0 * 4 + rA;
      v2f bv; bv.x = bp[0]; bv.y = bp[4];
      v8f cz = (v8f)0.0f;
      v8f d = __builtin_amdgcn_wmma_f32_16x16x4_f32(
          false, av, false, bv, (short)0, cz, false, false);
      // Scatter diagonal block: keep extract indices compile-time constant in
      // each divergent arm (predicated DS stores, no cndmask select chains).
      float* o = dst + (rightBase + selj) * 16 + colD;
      if (doStore && hiD) {
        o[0] = d[4]; o[4] = d[5]; o[8] = d[6]; o[12] = d[7];
      } else if (doStore) {
        o[0] = d[0]; o[4] = d[1]; o[8] = d[2]; o[12] = d[3];
      }
    }
  }
  __syncthreads();  // final scan (in scan0) complete

  // -- 4) exclusive carry, re-walk segment, emit masked positions ----------
  const int ci = (tid == 0) ? IDSLOT : (tid - 1);
  float C[12];
  {
    const float* xf = scan0 + ci * 16;
#pragma unroll
    for (int e = 0; e < 12; ++e) C[e] = xf[e];
  }
  const int vlim = 3 * lens[b];
  __syncthreads();  // carries read; coords may now overwrite scan region

  for (int rj = 0; rj < NRES; ++rj)
    do_residue<true>(C, tid * NRES + rj, phiS, psiS, ca0, sa0, ca1, sa1, ca2,
                     sa2, coords, vlim);
  __syncthreads();

  // -- 5) TDM: DMA the 9L-float coord tile to d_out + b*9L ------------------
  if (tid < 32) {
    const uint64_t ga = (uint64_t)(uintptr_t)(out + (size_t)b * (9 * L));
    const unsigned lo = (unsigned)(uintptr_t)(void*)coords;
    TdmDesc d = tdm_desc_1d(ga, lo, 9u * L);
    __builtin_amdgcn_tensor_store_from_lds(d.g0, d.g1, d.g2, d.g3, (v8i)0, 0);
    __builtin_amdgcn_s_wait_tensorcnt(0);
  }
}

extern "C" void kernel_launch(void* const* d_in, const int* in_sizes, int n_in,
                              void* d_out, int out_size, void* d_ws,
                              size_t ws_size, hipStream_t stream) {
  (void)n_in; (void)out_size; (void)d_ws; (void)ws_size;
  const float* inp = (const float*)d_in[0];
  const int* lens  = (const int*)d_in[1];
  float* out       = (float*)d_out;
  const int B = in_sizes[1];  // 256 chains, one workgroup each
  hipLaunchKernelGGL(angles2backbone_scan_kernel, dim3(B), dim3(256), 0, stream,
                     inp, lens, out);
}